// YOLOv3Head_84748294685095
// MI455X (gfx1250) — compile-verified
//
#include <hip/hip_runtime.h>
#include <stdint.h>

// ---------------------------------------------------------------------------
// YOLOv3 head decode for MI455X (gfx1250).
// Memory-bound (~250MB traffic -> ~11us floor @ 23.3TB/s). Strategy:
//  * One TDM tensor_load_to_lds per block stages the full strided
//    258ch x 85pos f32 tile (87.7KB) into LDS (tracked by TENSORcnt) —
//    ALL bulk input traffic goes through the Tensor Data Mover.
//  * wave32 threads t = hw*3 + a  ->  block's box/score outputs are a single
//    contiguous span; b128 non-temporal stores (footprint > 192MB L2).
//  * hardware v_exp_f32 transcendentals (__expf/__powf) keep VALU time well
//    under the memory time.
// ---------------------------------------------------------------------------

#define ENABLE_TDM 1

#define NUM_CLASSES 80
#define AN_NUM      3
#define C_IN        258        // 3 ioup + 3*85 anchor channels (contiguous)
#define NTOT        22743      // (361+1444+5776)*3
#define TILE_HW     85
#define BLOCK       256
#define EPS_F       1e-7f
#define BIG_F       1e7f
#define SCALE_XY    1.05f
#define IOU_FACTOR  0.4f

typedef float    v4f __attribute__((ext_vector_type(4)));
typedef uint32_t v4u __attribute__((ext_vector_type(4)));
typedef uint32_t v8u __attribute__((ext_vector_type(8)));

// anchors[level][a][0]=w, [1]=h
__device__ __constant__ float c_anchors[3][3][2] = {
    {{116.0f,  90.0f}, {156.0f, 198.0f}, {373.0f, 326.0f}},
    {{ 30.0f,  61.0f}, { 62.0f,  45.0f}, { 59.0f, 119.0f}},
    {{ 10.0f,  13.0f}, { 16.0f,  30.0f}, { 33.0f,  23.0f}}
};

__device__ __forceinline__ float fast_sigmoid(float x) {
    return 1.0f / (1.0f + __expf(-x));
}

__global__ __launch_bounds__(BLOCK)
void yolo_decode_kernel(const float* __restrict__ in0,
                        const float* __restrict__ in1,
                        const float* __restrict__ in2,
                        const float* __restrict__ imsz,
                        float* __restrict__ boxes,
                        float* __restrict__ scores)
{
    // 258 rows (channel 0..257) x cnt positions, row-major, rows packed
    // back-to-back with row stride == cnt (TDM writes the tile contiguously).
    __shared__ float smem[C_IN * TILE_HW];

    const int bx = blockIdx.x;
    const int b  = blockIdx.y;

    int level, tile, H, stride, n_off;
    const float* in;
    if (bx < 5)       { level = 0; tile = bx;      H = 19; stride = 32; n_off = 0;    in = in0; }
    else if (bx < 22) { level = 1; tile = bx - 5;  H = 38; stride = 16; n_off = 1083; in = in1; }
    else              { level = 2; tile = bx - 22; H = 76; stride = 8;  n_off = 5415; in = in2; }
    const int HW  = H * H;
    const int hw0 = tile * TILE_HW;
    const int cnt = min(TILE_HW, HW - hw0);      // uniform per block

    // Tile start in global memory: channel 0, position hw0.
    const float* gsrc = in + (size_t)b * C_IN * (size_t)HW + hw0;

#if ENABLE_TDM
    if (threadIdx.x == 0) {
        // ---- Tensor DMA descriptor (D#), 2-D tile: x = hw (contig), y = channel
        const uint64_t ga  = (uint64_t)(uintptr_t)gsrc;
        const uint32_t lds = (uint32_t)(uintptr_t)&smem[0];   // low 32b == LDS offset

        v4u g0;
        g0[0] = 1u;                                    // count=1, user desc
        g0[1] = lds;                                   // lds_addr
        g0[2] = (uint32_t)ga;                          // global_addr[31:0]
        g0[3] = (uint32_t)(ga >> 32) | 0x80000000u;    // global_addr[56:32] | type=2

        v8u g1;
        const uint32_t td0 = (uint32_t)cnt;            // tensor_dim0 (OOB bound in x)
        const uint32_t td1 = (uint32_t)C_IN;           // tensor_dim1 = 258 channels
        g1[0] = 2u << 16;                              // wg_mask=0, data_size=2 (4B)
        g1[1] = (td0 & 0xFFFFu) << 16;                 // tensor_dim0[15:0]
        g1[2] = (td0 >> 16) | ((td1 & 0xFFFFu) << 16); // tensor_dim0[31:16] | tensor_dim1[15:0]
        g1[3] = (td1 >> 16) | ((uint32_t)cnt << 16);   // tensor_dim1[31:16] | tile_dim0=cnt
        g1[4] = (uint32_t)C_IN;                        // tile_dim1=258, tile_dim2=0 (2-D)
        g1[5] = (uint32_t)HW;                          // tensor_dim0_stride lo32 (elements)
        g1[6] = 0u;                                    // stride0 hi16 | stride1 lo16 (unused)
        g1[7] = 0u;                                    // stride1 hi32 (unused)

        // 2-D tensor: descriptor groups 2/3 are NULL.
        asm volatile("tensor_load_to_lds %0, %1, null, null"
                     :: "s"(g0), "s"(g1)
                     : "memory");
        __builtin_amdgcn_s_wait_tensorcnt(0);
    }
    __syncthreads();
#else
    {
        const int total = C_IN * cnt;
        for (int i = threadIdx.x; i < total; i += BLOCK) {
            const int c = i / cnt;
            const int x = i - c * cnt;
            smem[i] = gsrc[(size_t)c * HW + x];
        }
        __syncthreads();
    }
#endif

    const int t = threadIdx.x;
    if (t >= 3 * cnt) return;                 // no barriers past this point

    const int hw_local = t / 3;
    const int a        = t - hw_local * 3;
    const int hw       = hw0 + hw_local;
    const int wcol     = hw % H;              // gx (W == H)
    const int hrow     = hw / H;              // gy

    // Staged logits for (this anchor, this position):
    //   ioup  -> row a
    //   dx..prob -> rows 3 + a*85 + c
    const float  ioupl = smem[(size_t)a * cnt + hw_local];
    const float* base  = smem + (size_t)(3 + a * 85) * cnt + hw_local;
    const float dx   = base[0];
    const float dy   = base[(size_t)1 * cnt];
    const float dw   = base[(size_t)2 * cnt];
    const float dh   = base[(size_t)3 * cnt];
    const float objl = base[(size_t)4 * cnt];

    const float imh = imsz[b * 2 + 0];
    const float imw = imsz[b * 2 + 1];
    const float inv = 1.0f / ((float)H * (float)stride);   // == 1/608 every level
    const float fx  = imw * inv;
    const float fy  = imh * inv;

    // conf = sigmoid(de_sigmoid(obj^0.6 * ip^0.4)) == 1/(1+clamp(1/clamp(x)-1))
    const float obj  = fast_sigmoid(objl);
    const float ip   = fast_sigmoid(ioupl);
    const float nobj = __powf(obj, 1.0f - IOU_FACTOR) * __powf(ip, IOU_FACTOR);
    float xcl = fminf(fmaxf(nobj, EPS_F), BIG_F);
    float tt  = 1.0f / xcl - 1.0f;
    tt = fminf(fmaxf(tt, EPS_F), BIG_F);
    const float conf = 1.0f / (1.0f + tt);

    // boxes
    const float xc = (SCALE_XY * fast_sigmoid(dx) + (float)wcol - (SCALE_XY - 1.0f) * 0.5f) * (float)stride;
    const float yc = (SCALE_XY * fast_sigmoid(dy) + (float)hrow - (SCALE_XY - 1.0f) * 0.5f) * (float)stride;
    const float aw = c_anchors[level][a][0];
    const float ah = c_anchors[level][a][1];
    const float wx = __expf(dw) * aw;
    const float wy = __expf(dh) * ah;

    float x0 = (xc - 0.5f * wx) * fx;
    float y0 = (yc - 0.5f * wy) * fy;
    float x1 = (xc + 0.5f * wx) * fx;
    float y1 = (yc + 0.5f * wy) * fy;
    x0 = (x0 < 0.0f) ? 0.0f : x0;
    y0 = (y0 < 0.0f) ? 0.0f : y0;
    x1 = (x1 > imw) ? imw : x1;
    y1 = (y1 > imh) ? imh : y1;

    const int    n   = n_off + hw * 3 + a;
    const size_t pos = (size_t)b * NTOT + n;

    v4f bb; bb[0] = x0; bb[1] = y0; bb[2] = x1; bb[3] = y1;
    __builtin_nontemporal_store(bb, (v4f*)boxes + pos);    // 16B aligned

    // scores: 80 contiguous floats; block's span is fully contiguous
    v4f* sdst = (v4f*)(scores + pos * NUM_CLASSES);
#pragma unroll
    for (int c4 = 0; c4 < NUM_CLASSES / 4; ++c4) {
        v4f sv;
#pragma unroll
        for (int j = 0; j < 4; ++j) {
            const float pl = base[(size_t)(5 + c4 * 4 + j) * cnt];
            sv[j] = conf * fast_sigmoid(pl);
        }
        __builtin_nontemporal_store(sv, sdst + c4);
    }
}

extern "C" void kernel_launch(void* const* d_in, const int* in_sizes, int n_in,
                              void* d_out, int out_size, void* d_ws, size_t ws_size,
                              hipStream_t stream) {
    const float* in0  = (const float*)d_in[0];   // (B,258,19,19)
    const float* in1  = (const float*)d_in[1];   // (B,258,38,38)
    const float* in2  = (const float*)d_in[2];   // (B,258,76,76)
    const float* imsz = (const float*)d_in[3];   // (B,2)

    const int B = in_sizes[3] / 2;

    float* boxes  = (float*)d_out;
    float* scores = boxes + (size_t)B * NTOT * 4;

    // 90 tile-blocks cover all three levels: 5 (19x19) + 17 (38x38) + 68 (76x76)
    dim3 grid(90, B);
    yolo_decode_kernel<<<grid, BLOCK, 0, stream>>>(in0, in1, in2, imsz, boxes, scores);
}